// W4O16Embedding_40243843564270
// MI455X (gfx1250) — compile-verified
//
#include <hip/hip_runtime.h>

// W4 (int4-packed) embedding gather + dequant to fp16.
// CDNA5 path: async global->LDS row staging (ASYNCcnt) + packed fp16 VALU
// dequant + non-temporal b128 streaming stores.

typedef _Float16 half_t;
typedef _Float16 h2   __attribute__((ext_vector_type(2)));
typedef unsigned int  u32;
typedef u32      u32x4 __attribute__((ext_vector_type(4)));

#define TOKENS_PER_BLOCK 8    // one token per wave32, 8 waves per block
#define ROW_I32          512  // EMBED_DIM(4096) / PACK(8)
#define EMBED_DIM        4096

__global__ __launch_bounds__(256)
void w4_embed_dequant_kernel(const int* __restrict__ x,
                             const int* __restrict__ weight,
                             const half_t* __restrict__ scales,
                             half_t* __restrict__ out,
                             int n_tok)
{
    __shared__ int lds[TOKENS_PER_BLOCK * ROW_I32]; // 8 rows * 2KB = 16KB

    const int lane = threadIdx.x & 31;
    const int wave = threadIdx.x >> 5;
    const int g    = blockIdx.x * TOKENS_PER_BLOCK + wave;
    if (g >= n_tok) return;   // uniform per wave

    // Token id is wave-uniform -> force scalar so the row base lives in SGPRs
    // (feeds the async-load SADDR operand directly).
    int tok = __builtin_amdgcn_readfirstlane(x[g]);

    const int* rowBase = weight + (size_t)tok * ROW_I32;

    // Per-lane LDS byte address and per-lane global byte offset (b128 chunks).
    // Generic->int truncation yields the wave-relative LDS byte offset.
    unsigned ldsAddr = (unsigned)(size_t)(&lds[wave * ROW_I32]) + (unsigned)(lane * 16);
    unsigned vOff    = (unsigned)(lane * 16);

    // Stage the 2048B packed row into LDS with 4 async b128 transfers.
    // The same immediate offset is applied to both global and LDS addresses,
    // so one base pair covers the whole row. Tracked by ASYNCcnt.
    asm volatile(
        "global_load_async_to_lds_b128 %0, %1, %2\n\t"
        "global_load_async_to_lds_b128 %0, %1, %2 offset:512\n\t"
        "global_load_async_to_lds_b128 %0, %1, %2 offset:1024\n\t"
        "global_load_async_to_lds_b128 %0, %1, %2 offset:1536"
        :
        : "v"(ldsAddr), "v"(vOff), "s"(rowBase)
        : "memory");

    // Overlap: fetch the per-row scale while the async DMA is in flight.
    half_t s  = scales[tok];
    h2 s2     = {s, s};
    h2 bias   = {(half_t)1032.0f, (half_t)1032.0f}; // 1024 (magic) + 8 (zero point)

    half_t* outRow = out + (size_t)g * EMBED_DIM;

    // Drain the async gather before touching LDS.
    asm volatile("s_wait_asynccnt 0x0" ::: "memory");

    const int* myLds = &lds[wave * ROW_I32];

    // Lane handles int32 positions p = lane + k*32:
    //  - ds_load_b32 stride 128B  -> zero bank conflicts
    //  - each int32 expands to 8 fp16 = 16B at out offset p*16
    //    -> lanes write contiguous 16B chunks: perfectly coalesced b128 stores.
    #pragma unroll
    for (int k = 0; k < 16; ++k) {
        const int p = lane + k * 32;
        const u32 w = (u32)myLds[p];

        // Nibble interleave: elem 2m   = nibble m      (bits 4m+3 : 4m)
        //                    elem 2m+1 = nibble m+4    (bits 4m+19: 4m+16)
        // OR into fp16 magic 0x6400 => value 1024+n exactly.
        union { h2 h; u32 u; } q0, q1, q2, q3, r0, r1, r2, r3;
        q0.u = ( w         & 0x000F000Fu) | 0x64006400u; // elems 0,1
        q1.u = ((w >>  4)  & 0x000F000Fu) | 0x64006400u; // elems 2,3
        q2.u = ((w >>  8)  & 0x000F000Fu) | 0x64006400u; // elems 4,5
        q3.u = ((w >> 12)  & 0x000F000Fu) | 0x64006400u; // elems 6,7

        // ((1024+n) - 1032) * s == (n - 8) * s, bit-exact vs reference fp16 math
        // (subtraction exact, single rounding in the packed multiply).
        r0.h = (q0.h - bias) * s2;
        r1.h = (q1.h - bias) * s2;
        r2.h = (q2.h - bias) * s2;
        r3.h = (q3.h - bias) * s2;

        u32x4 o = {r0.u, r1.u, r2.u, r3.u};
        // 256MB output stream > 192MB L2: bypass with non-temporal store.
        __builtin_nontemporal_store(o, (u32x4*)(outRow + (size_t)p * 8));
    }
}

extern "C" void kernel_launch(void* const* d_in, const int* in_sizes, int n_in,
                              void* d_out, int out_size, void* d_ws, size_t ws_size,
                              hipStream_t stream) {
    const int*    x      = (const int*)d_in[0];      // [4*8192] int32 token ids
    const int*    weight = (const int*)d_in[1];      // [128000, 512] int32 packed
    const half_t* scales = (const half_t*)d_in[2];   // [128000] fp16
    half_t*       out    = (half_t*)d_out;           // [4*8192, 4096] fp16

    const int n_tok  = in_sizes[0];
    const int blocks = (n_tok + TOKENS_PER_BLOCK - 1) / TOKENS_PER_BLOCK;

    hipLaunchKernelGGL(w4_embed_dequant_kernel,
                       dim3(blocks), dim3(256), 0, stream,
                       x, weight, scales, out, n_tok);
}